// RTDETRPostProcessor_61649960567347
// MI455X (gfx1250) — compile-verified
//
#include <hip/hip_runtime.h>
#include <hip/hip_bf16.h>
#include <math.h>

typedef float v4f __attribute__((ext_vector_type(4)));

#define B_   2
#define Q_   300
#define C_   80
#define K_   300
#define NQC  (Q_ * C_)     // 24000 flattened scores per batch
#define HM   128
#define OUTS 480
#define CAND 512

// ---- monotonic float -> sortable uint mapping --------------------------------
__device__ __forceinline__ unsigned key_of(float f) {
  unsigned u = __float_as_uint(f);
  return (u & 0x80000000u) ? ~u : (u | 0x80000000u);
}

// =============================================================================
// Kernel A: per-batch top-300 (radix-select + bitonic sort of candidates),
// writes scores / labels / scaled xyxy boxes, and query indices to workspace.
// grid = B blocks of 1024 threads.
// Output layout (floats): [0,600) scores | [600,1200) labels |
//                         [1200,3600) boxes | [3600, ...) masks
// =============================================================================
__global__ __launch_bounds__(1024)
void topk_boxes_kernel(const float* __restrict__ logits,
                       const float* __restrict__ boxes,
                       const float* __restrict__ sizes,
                       float* __restrict__ out,
                       int* __restrict__ qidx) {
  const int b = blockIdx.x;
  const int t = threadIdx.x;
  const float* lg = logits + (size_t)b * NQC;

  __shared__ unsigned hist[16];
  __shared__ unsigned sh_pref;
  __shared__ unsigned sh_rem;
  __shared__ unsigned sh_cnt;
  __shared__ unsigned long long cand[CAND];

  if (t == 0) { sh_pref = 0u; sh_rem = (unsigned)K_; }
  __syncthreads();

  // ---- 8-pass nibble radix-select for the K-th largest key -----------------
  for (int p = 0; p < 8; ++p) {
    const int shift = 28 - 4 * p;
    const unsigned pmask = p ? (0xFFFFFFFFu << (shift + 4)) : 0u;
    if (t < 16) hist[t] = 0u;
    __syncthreads();
    const unsigned pref = sh_pref;
    for (int i = t; i < NQC; i += 1024) {
      unsigned k = key_of(lg[i]);
      if ((k & pmask) == pref) atomicAdd(&hist[(k >> shift) & 15u], 1u);
    }
    __syncthreads();
    if (t == 0) {
      unsigned rem = sh_rem, cum = 0u, d = 0u;
      for (int dd = 15; dd >= 0; --dd) {
        unsigned h = hist[dd];
        if (cum + h >= rem) { d = (unsigned)dd; break; }
        cum += h;
      }
      sh_pref = pref | (d << shift);
      sh_rem  = rem - cum;
    }
    __syncthreads();
  }

  // ---- collect all candidates with key >= T --------------------------------
  const unsigned T = sh_pref;
  if (t < CAND) cand[t] = 0ull;
  if (t == 0) sh_cnt = 0u;
  __syncthreads();
  for (int i = t; i < NQC; i += 1024) {
    unsigned k = key_of(lg[i]);
    if (k >= T) {
      unsigned pos = atomicAdd(&sh_cnt, 1u);
      if (pos < CAND)
        cand[pos] = ((unsigned long long)k << 32) |
                    (unsigned long long)(0xFFFFFFFFu - (unsigned)i);
    }
  }
  __syncthreads();

  // ---- bitonic sort, descending on (key desc, index asc) composite ---------
  for (unsigned sz = 2; sz <= CAND; sz <<= 1) {
    for (unsigned st = sz >> 1; st > 0; st >>= 1) {
      __syncthreads();
      if (t < CAND) {
        unsigned j = (unsigned)t ^ st;
        if (j > (unsigned)t) {
          unsigned long long a = cand[t], c2 = cand[j];
          bool desc = (((unsigned)t & sz) == 0u);
          if (desc ? (a < c2) : (a > c2)) { cand[t] = c2; cand[j] = a; }
        }
      }
    }
  }
  __syncthreads();

  // ---- emit scores / labels / boxes / query indices ------------------------
  if (t < K_) {
    unsigned long long cv = cand[t];
    unsigned k   = (unsigned)(cv >> 32);
    unsigned idx = 0xFFFFFFFFu - (unsigned)(cv & 0xFFFFFFFFull);
    unsigned u   = (k & 0x80000000u) ? (k ^ 0x80000000u) : ~k;
    float logit  = __uint_as_float(u);
    float score  = 1.0f / (1.0f + expf(-logit));
    unsigned label = idx % (unsigned)C_;
    unsigned q     = idx / (unsigned)C_;

    out[b * K_ + t]            = score;
    out[B_ * K_ + b * K_ + t]  = (float)label;

    const float* bx = boxes + ((size_t)b * Q_ + q) * 4;
    float cx = bx[0], cy = bx[1], w = bx[2], h = bx[3];
    float s0 = sizes[b * 2 + 0], s1 = sizes[b * 2 + 1];
    float* ob = out + 2 * B_ * K_ + ((size_t)b * K_ + t) * 4;
    ob[0] = (cx - 0.5f * w) * s0;
    ob[1] = (cy - 0.5f * h) * s1;
    ob[2] = (cx + 0.5f * w) * s0;
    ob[3] = (cy + 0.5f * h) * s1;

    qidx[b * K_ + t] = (int)q;
  }
}

// =============================================================================
// Kernel B: per selected mask: async-DMA 128x128 logits -> LDS (overlapped with
// building the separable interpolation tables), sigmoid in place, bilinear
// upsample to 480x480 (half-pixel, edge-clamp) via LDS tables, threshold,
// non-temporal b128 stores. grid = B*K blocks of 256 threads.
// =============================================================================
__global__ __launch_bounds__(256)
void mask_resize_kernel(const float* __restrict__ masks,
                        const int* __restrict__ qidx,
                        float* __restrict__ out) {
  const int bk = blockIdx.x;           // 0 .. B*K-1
  const int b  = bk / K_;
  const int t  = threadIdx.x;
  __shared__ __align__(16) float m[HM * HM];   // 64 KB tile
  __shared__ int   tp[OUTS];                   // packed x0 | (x1 << 16)
  __shared__ float tw[OUTS];                   // lerp weight

  const int q = qidx[bk];
  const float* src = masks + ((size_t)b * Q_ + (size_t)q) * (HM * HM);
  unsigned long long sp = (unsigned long long)(const void*)src;   // SGPR pair base
  unsigned lbase = (unsigned)(size_t)(void*)&m[0];                // LDS byte offset

  // ---- CDNA5 async global->LDS DMA of the 64KB tile (ASYNCcnt path) --------
  for (int i = t; i < (HM * HM) / 4; i += 256) {
    unsigned lo = lbase + (unsigned)(i * 16);
    unsigned go = (unsigned)(i * 16);
    asm volatile("global_load_async_to_lds_b128 %0, %1, %2"
                 :: "v"(lo), "v"(go), "s"(sp) : "memory");
  }

  // ---- build separable interp tables while the DMA is in flight ------------
  const float s   = (float)HM / (float)OUTS;   // 128/480
  const float off = 0.5f * s - 0.5f;
  for (int o = t; o < OUTS; o += 256) {
    float f  = fmaf((float)o, s, off);
    float fl = floorf(f);
    int   i0 = (int)fl;
    int   x0 = min(max(i0, 0), HM - 1);
    int   x1 = min(max(i0 + 1, 0), HM - 1);
    tp[o] = x0 | (x1 << 16);
    tw[o] = f - fl;
  }

  asm volatile("s_wait_asynccnt 0x0" ::: "memory");

  // sigmoid in place; each wave touches exactly the slots it DMA'd
  for (int i = t; i < (HM * HM) / 4; i += 256) {
    v4f v = *(const v4f*)&m[i * 4];
    v.x = 1.0f / (1.0f + __expf(-v.x));
    v.y = 1.0f / (1.0f + __expf(-v.y));
    v.z = 1.0f / (1.0f + __expf(-v.z));
    v.w = 1.0f / (1.0f + __expf(-v.w));
    *(v4f*)&m[i * 4] = v;
  }
  __syncthreads();

  float* dst = out + 2 * B_ * K_ + 4 * B_ * K_ +  // = 3600
               (size_t)bk * ((size_t)OUTS * OUTS);

  for (int si = t; si < OUTS * (OUTS / 4); si += 256) {
    int Y = si / (OUTS / 4);
    int X = (si % (OUTS / 4)) * 4;

    int   py = tp[Y];
    float wy = tw[Y];
    const float* r0 = &m[(py & 0xFFFF) * HM];
    const float* r1 = &m[(py >> 16) * HM];

    v4f o;
#pragma unroll
    for (int c = 0; c < 4; ++c) {
      int   px = tp[X + c];
      float wx = tw[X + c];
      int x0 = px & 0xFFFF;
      int x1 = px >> 16;
      float a  = r0[x0], b2 = r0[x1];
      float c2 = r1[x0], d2 = r1[x1];
      float top = fmaf(wx, b2 - a, a);
      float bot = fmaf(wx, d2 - c2, c2);
      float v   = fmaf(wy, bot - top, top);
      o[c] = (v > 0.5f) ? 1.0f : 0.0f;
    }
    // 553MB of output >> 192MB L2: bypass with non-temporal b128 stores
    __builtin_nontemporal_store(o, (v4f*)(dst + (size_t)Y * OUTS + X));
  }
}

// =============================================================================
extern "C" void kernel_launch(void* const* d_in, const int* in_sizes, int n_in,
                              void* d_out, int out_size, void* d_ws, size_t ws_size,
                              hipStream_t stream) {
  const float* logits = (const float*)d_in[0];  // [2,300,80]
  const float* boxes  = (const float*)d_in[1];  // [2,300,4]
  const float* masks  = (const float*)d_in[2];  // [2,300,128,128]
  const float* sizes  = (const float*)d_in[3];  // [2,2]
  float* out = (float*)d_out;
  int* qidx  = (int*)d_ws;                      // B*K ints of scratch

  topk_boxes_kernel<<<B_, 1024, 0, stream>>>(logits, boxes, sizes, out, qidx);
  mask_resize_kernel<<<B_ * K_, 256, 0, stream>>>(masks, qidx, out);

  (void)in_sizes; (void)n_in; (void)out_size; (void)ws_size;
}